// SpikeLoss_41308995453605
// MI455X (gfx1250) — compile-verified
//
#include <hip/hip_runtime.h>

// ---------------------------------------------------------------------------
// SpikeLoss: 0.5 * sum((outputs - psp(target))^2)
//   psp: syn_t = syn_{t-1} * (1 - 1/tau) + x_t ; emit syn_t / tau   (tau = 5)
// Shape [B,C,H,W,T] = [16,128,16,16,100]; T is innermost/contiguous.
// Memory-bound: ~419 MB read -> ~18 us at 23.3 TB/s. Strategy: stage target
// tiles into LDS with CDNA5 async global->LDS DMA (coalesced 512B/instr),
// run the serial time-scan per lane out of LDS, stream `outputs` coalesced,
// deterministic two-stage tree reduction (no float atomics).
// ---------------------------------------------------------------------------

#define T_STEPS 100

constexpr int   kPixPerWave   = 32;
constexpr int   kWavesPerBlk  = 4;
constexpr int   kThreads      = kWavesPerBlk * 32;           // 128
constexpr int   kPixPerBlock  = kPixPerWave * kWavesPerBlk;  // 128
constexpr int   kF4PerWave    = kPixPerWave * T_STEPS / 4;   // 800 float4 / wave tile
constexpr int   kF4Iters      = kF4PerWave / 32;             // 25 iterations
constexpr float kDecay        = 0.8f;                        // 1 - 1/tau
constexpr float kInvTau       = 0.2f;                        // 1/tau

#if defined(__has_builtin)
#  if __has_builtin(__builtin_amdgcn_global_load_async_to_lds_b128)
#    define USE_ASYNC_LDS 1
#  else
#    define USE_ASYNC_LDS 0
#  endif
#else
#  define USE_ASYNC_LDS 0
#endif

// Builtin prototype (from hipcc diagnostic):
//   void __builtin_amdgcn_global_load_async_to_lds_b128(
//       int4 addrspace(1)* src, int4 addrspace(3)* dst, imm int off, imm int cpol)
typedef int v4i __attribute__((ext_vector_type(4)));
typedef __attribute__((address_space(1))) v4i Gv4i;
typedef __attribute__((address_space(3))) v4i Lv4i;

__device__ __forceinline__ void wait_async_zero() {
#if defined(__has_builtin) && __has_builtin(__builtin_amdgcn_s_wait_asynccnt)
  __builtin_amdgcn_s_wait_asynccnt(0);
#else
  asm volatile("s_wait_asynccnt 0" ::: "memory");
#endif
}

__global__ __launch_bounds__(kThreads)
void spike_loss_partial(const float* __restrict__ outputs,
                        const float* __restrict__ target,
                        float* __restrict__ partial,
                        int nPix) {
  // 4 waves x (32 pixels x 100 t) x 4B = 51200 B static LDS
  __shared__ float lds[kWavesPerBlk][kPixPerWave * T_STEPS];
  __shared__ float wred[kWavesPerBlk];

  const int tid  = threadIdx.x;
  const int wave = tid >> 5;
  const int lane = tid & 31;
  const int pixelBase = (blockIdx.x * kWavesPerBlk + wave) * kPixPerWave;

  float sum = 0.0f;

  if (pixelBase + kPixPerWave <= nPix) {
    float* ldsW = &lds[wave][0];
    const float* tgtBase = target  + (size_t)pixelBase * T_STEPS;
    const float* outBase = outputs + (size_t)pixelBase * T_STEPS;

    // ---- Phase 1: stage 12.8 KB of target into LDS (flat copy, coalesced) --
#if USE_ASYNC_LDS
    #pragma unroll
    for (int k = 0; k < kF4Iters; ++k) {
      const int q = k * 32 + lane;  // float4 index within wave tile
      __builtin_amdgcn_global_load_async_to_lds_b128(
          (Gv4i*)(tgtBase + 4 * q),
          (Lv4i*)(ldsW + 4 * q),
          /*imm offset*/ 0, /*cpol*/ 0);
    }
    wait_async_zero();
#else
    #pragma unroll
    for (int k = 0; k < kF4Iters; ++k) {
      const int q = k * 32 + lane;
      const float4 v = *reinterpret_cast<const float4*>(tgtBase + 4 * q);
      *reinterpret_cast<float4*>(ldsW + 4 * q) = v;
    }
#endif
    __builtin_amdgcn_wave_barrier();
    asm volatile("" ::: "memory");

    // ---- Phase 2: serial leaky-integrator scan, lane l owns pixel l -------
    {
      float* row = ldsW + lane * T_STEPS;
      float syn = 0.0f;
      #pragma unroll 10
      for (int t = 0; t < T_STEPS; ++t) {
        syn = fmaf(syn, kDecay, row[t]);
        row[t] = syn * kInvTau;  // psp overwrites target in LDS
      }
    }
    __builtin_amdgcn_wave_barrier();
    asm volatile("" ::: "memory");

    // ---- Phase 3: stream outputs coalesced, accumulate squared error ------
    #pragma unroll
    for (int k = 0; k < kF4Iters; ++k) {
      const int q = k * 32 + lane;
      const float4 o = *reinterpret_cast<const float4*>(outBase + 4 * q);
      const float d0 = o.x - ldsW[4 * q + 0];
      const float d1 = o.y - ldsW[4 * q + 1];
      const float d2 = o.z - ldsW[4 * q + 2];
      const float d3 = o.w - ldsW[4 * q + 3];
      sum = fmaf(d0, d0, sum);
      sum = fmaf(d1, d1, sum);
      sum = fmaf(d2, d2, sum);
      sum = fmaf(d3, d3, sum);
    }
  } else if (pixelBase < nPix) {
    // Ragged tail (not hit for the reference shape): safe scalar path.
    const int p = pixelBase + lane;
    if (p < nPix) {
      const float* tp = target  + (size_t)p * T_STEPS;
      const float* op = outputs + (size_t)p * T_STEPS;
      float syn = 0.0f;
      for (int t = 0; t < T_STEPS; ++t) {
        syn = fmaf(syn, kDecay, tp[t]);
        const float d = op[t] - syn * kInvTau;
        sum = fmaf(d, d, sum);
      }
    }
  }

  // ---- Wave32 tree reduction -----------------------------------------------
  #pragma unroll
  for (int off = 16; off > 0; off >>= 1)
    sum += __shfl_xor(sum, off, 32);

  if (lane == 0) wred[wave] = sum;
  __syncthreads();
  if (tid == 0) {
    float s = 0.0f;
    #pragma unroll
    for (int w = 0; w < kWavesPerBlk; ++w) s += wred[w];
    partial[blockIdx.x] = s;  // fixed order -> deterministic
  }
}

__global__ __launch_bounds__(256)
void spike_loss_final(const float* __restrict__ partial, int n,
                      float* __restrict__ out) {
  __shared__ double s[256];
  double acc = 0.0;
  for (int i = threadIdx.x; i < n; i += 256) acc += (double)partial[i];
  s[threadIdx.x] = acc;
  __syncthreads();
  #pragma unroll
  for (int off = 128; off > 0; off >>= 1) {
    if (threadIdx.x < off) s[threadIdx.x] += s[threadIdx.x + off];
    __syncthreads();
  }
  if (threadIdx.x == 0) out[0] = (float)(0.5 * s[0]);
}

extern "C" void kernel_launch(void* const* d_in, const int* in_sizes, int n_in,
                              void* d_out, int out_size, void* d_ws, size_t ws_size,
                              hipStream_t stream) {
  const float* outputs = (const float*)d_in[0];
  const float* target  = (const float*)d_in[1];
  float* out = (float*)d_out;
  float* partial = (float*)d_ws;

  const int n    = in_sizes[0];          // B*C*H*W*T = 52,428,800
  const int nPix = n / T_STEPS;          // 524,288 pixels
  const int nBlocks = (nPix + kPixPerBlock - 1) / kPixPerBlock;  // 4096

  spike_loss_partial<<<nBlocks, kThreads, 0, stream>>>(outputs, target,
                                                       partial, nPix);
  spike_loss_final<<<1, 256, 0, stream>>>(partial, nBlocks, out);
}